// DiscreteAutoregressiveFlow_7868380086668
// MI455X (gfx1250) — compile-verified
//
#include <hip/hip_runtime.h>

typedef float v2f __attribute__((ext_vector_type(2)));
typedef float v8f __attribute__((ext_vector_type(8)));

#define VOCAB 64
#define WAVES_PER_WG 4
#define TOKENS_PER_WAVE 16
#define NET_PAD 130          // 16 rows * 130 floats = 2080 floats per wave region
#define WAVE_LDS 2080        // union: net[16][130]  OR  acc[2][16][65]

__global__ __launch_bounds__(WAVES_PER_WG * 32)
void daf_forward_kernel(const float* __restrict__ inp,
                        const float* __restrict__ W,
                        const float* __restrict__ bias,
                        float* __restrict__ out,
                        int m_total)
{
    __shared__ float lds[WAVES_PER_WG * WAVE_LDS];

    const int lane = threadIdx.x & 31;
    const int wave = threadIdx.x >> 5;
    const int wave_global = blockIdx.x * WAVES_PER_WG + wave;
    const int tok0 = wave_global * TOKENS_PER_WAVE;
    if (tok0 + TOKENS_PER_WAVE > m_total) return;   // wave-uniform guard

    const int m  = lane & 15;        // token row within tile / N column within tile
    const int kb = (lane >> 4) * 2;  // K sub-base: lanes 0-15 -> K{0,1}, 16-31 -> K{2,3}
    float* wlds = lds + wave * WAVE_LDS;

    // ---- Phase 1: A fragments: 16 tokens x 64 K, f32 (exact one-hot rows) ----
    // ISA 32-bit A 16x4 layout: lane m holds (K=kb, K=kb+1) for row M=m.
    v2f a[16];
    const float* arow = inp + (size_t)(tok0 + m) * VOCAB + kb;
#pragma unroll
    for (int k = 0; k < 16; ++k) {
        a[k] = *(const v2f*)(arow + 4 * k);
    }

    // ---- Phase 2: net = inputs @ W + b via chained V_WMMA_F32_16X16X4_F32 ----
    // 8 N-tiles of 16 columns (t=0..3 -> loc logits, t=4..7 -> scale logits)
#pragma unroll
    for (int t = 0; t < 8; ++t) {
        const int n0 = t * 16;
        const float bv = bias[n0 + m];     // same for every accumulator row
        v8f acc;
#pragma unroll
        for (int i = 0; i < 8; ++i) acc[i] = bv;

        // B 4x16 layout: VGPR0 = K=kb row, VGPR1 = K=kb+1 row, N = lane&15
        const float* wb = W + (size_t)kb * 128 + n0 + m;
#pragma unroll
        for (int k = 0; k < 16; ++k) {
            const float* wp = wb + (size_t)(4 * k) * 128;
            v2f bfrag;
            bfrag.x = wp[0];
            bfrag.y = wp[128];
            acc = __builtin_amdgcn_wmma_f32_16x16x4_f32(
                false, a[k], false, bfrag, (short)0, acc, false, false);
        }

        // D layout: VGPR r -> row (lane<16 ? r : r+8), col = n0 + (lane&15)
        const int rbase = (lane < 16) ? 0 : 8;
#pragma unroll
        for (int r = 0; r < 8; ++r) {
            wlds[(rbase + r) * NET_PAD + n0 + m] = acc[r];
        }
    }

    // ---- Phase 3: per-token argmax (first-max wins, matching jnp.argmax) ----
    // lanes 0-15: loc (cols 0..63) for token m; lanes 16-31: scale (cols 64..127)
    const int h = lane >> 4;
    float best = -__builtin_huge_valf();
    int idx = 0;
    const int abase = m * NET_PAD + h * VOCAB;
    for (int j = 0; j < VOCAB; ++j) {
        const float x = wlds[abase + j];
        if (x > best) { best = x; idx = j; }
    }
    const int c_l = __shfl(idx, m);        // loc argmax for my token
    const int c_s = __shfl(idx, m + 16);   // scale argmax for my token

    // ---- Phase 4: fused modular multiply + rotate as a scatter-add ----------
    //   out[(c_s*v + c_l) % 64] += inputs[v], gated by (c_s != 0)
    // Reuse the wave's LDS region as acc[2][16][65]; LDS is in-order per wave,
    // so the argmax loads above are safely ordered before these zero-stores.
    for (int i = lane; i < WAVE_LDS; i += 32) wlds[i] = 0.0f;

    const float gate = (c_s != 0) ? 1.0f : 0.0f;
    const float* irow = inp + (size_t)(tok0 + m) * VOCAB;
    const int accbase = h * (16 * 65) + m * 65;   // each lane owns its region
    for (int vv = 0; vv < 32; ++vv) {
        const int v = h * 32 + vv;
        const float x = irow[v] * gate;
        const int u = (c_s * v + c_l) & 63;
        wlds[accbase + u] += x;
    }

    // ---- Phase 5: combine halves, coalesced store of 16x64 outputs ----------
    float* orow = out + (size_t)tok0 * VOCAB;
    for (int i = lane; i < TOKENS_PER_WAVE * VOCAB; i += 32) {
        const int tr = i >> 6;
        const int u  = i & 63;
        orow[i] = wlds[tr * 65 + u] + wlds[16 * 65 + tr * 65 + u];
    }
}

extern "C" void kernel_launch(void* const* d_in, const int* in_sizes, int n_in,
                              void* d_out, int out_size, void* d_ws, size_t ws_size,
                              hipStream_t stream) {
    const float* inp = (const float*)d_in[0];   // [B, L, 64] f32 one-hot
    const float* W   = (const float*)d_in[1];   // [64, 128] f32
    const float* b   = (const float*)d_in[2];   // [128] f32
    float* out = (float*)d_out;                  // [B, L, 64] f32

    const int m_total = in_sizes[0] / VOCAB;     // B*L tokens
    const int waves   = (m_total + TOKENS_PER_WAVE - 1) / TOKENS_PER_WAVE;
    const int blocks  = (waves + WAVES_PER_WG - 1) / WAVES_PER_WG;

    daf_forward_kernel<<<blocks, WAVES_PER_WG * 32, 0, stream>>>(inp, W, b, out, m_total);
}